// TAGNet_bench_1769526526169
// MI455X (gfx1250) — compile-verified
//
#include <hip/hip_runtime.h>
#include <math.h>

typedef __attribute__((ext_vector_type(2))) float v2f;
typedef __attribute__((ext_vector_type(4))) float v4f;
typedef __attribute__((ext_vector_type(8))) float v8f;
typedef __attribute__((ext_vector_type(4))) int   v4i;

#define NT_LOAD(p) __builtin_nontemporal_load(p)

// ============ degree: deg[dst] += w  (quad-vectorized edge stream) ============
__global__ void deg_kernel4(const v4i* __restrict__ dst4, const v4f* __restrict__ w4,
                            float* __restrict__ deg, int Q) {
    int q = blockIdx.x * blockDim.x + threadIdx.x;
    if (q < Q) {
        v4i d = NT_LOAD(dst4 + q);          // global_load_b128 (NT)
        v4f w = NT_LOAD(w4 + q);
        unsafeAtomicAdd(deg + d.x, w.x);
        unsafeAtomicAdd(deg + d.y, w.y);
        unsafeAtomicAdd(deg + d.z, w.z);
        unsafeAtomicAdd(deg + d.w, w.w);
    }
}
__global__ void deg_kernel1(const int* __restrict__ dst, const float* __restrict__ w,
                            float* __restrict__ deg, int E) {
    int e = blockIdx.x * blockDim.x + threadIdx.x;
    if (e < E) unsafeAtomicAdd(deg + NT_LOAD(dst + e), NT_LOAD(w + e));
}

// ================= dis = deg>0 ? rsqrt(deg) : 0  (in place) ==================
__global__ void rsqrt_kernel(float* __restrict__ degdis, int N) {
    int i = blockIdx.x * blockDim.x + threadIdx.x;
    if (i < N) {
        float d = degdis[i];
        degdis[i] = (d > 0.f) ? rsqrtf(d) : 0.f;
    }
}

// ========== hop edge pass, F=1:  t[dst] += w * g[src]  (1 gather/edge) =======
__global__ void hop1_kernel4(const v4i* __restrict__ src4, const v4i* __restrict__ dst4,
                             const v4f* __restrict__ w4, const float* __restrict__ g,
                             float* __restrict__ t, int Q) {
    int q = blockIdx.x * blockDim.x + threadIdx.x;
    if (q < Q) {
        v4i s = NT_LOAD(src4 + q);
        v4i d = NT_LOAD(dst4 + q);
        v4f w = NT_LOAD(w4 + q);
        unsafeAtomicAdd(t + d.x, w.x * g[s.x]);   // g: L2-resident (2 MB)
        unsafeAtomicAdd(t + d.y, w.y * g[s.y]);
        unsafeAtomicAdd(t + d.z, w.z * g[s.z]);
        unsafeAtomicAdd(t + d.w, w.w * g[s.w]);
    }
}
__global__ void hop1_kernel1(const int* __restrict__ src, const int* __restrict__ dst,
                             const float* __restrict__ w, const float* __restrict__ g,
                             float* __restrict__ t, int E) {
    int e = blockIdx.x * blockDim.x + threadIdx.x;
    if (e < E) unsafeAtomicAdd(t + NT_LOAD(dst + e), NT_LOAD(w + e) * g[NT_LOAD(src + e)]);
}

// ========== hop edge pass, F=4:  t[dst,:] += w * g[src,:] ====================
__global__ void hop4_kernel4(const v4i* __restrict__ src4, const v4i* __restrict__ dst4,
                             const v4f* __restrict__ w4, const v4f* __restrict__ g,
                             float* __restrict__ t, int Q) {
    int q = blockIdx.x * blockDim.x + threadIdx.x;
    if (q < Q) {
        v4i s = NT_LOAD(src4 + q);
        v4i d = NT_LOAD(dst4 + q);
        v4f w = NT_LOAD(w4 + q);
        v4f h;
        float* o;
        h = g[s.x]; o = t + (size_t)d.x * 4;      // b128 gather from L2
        unsafeAtomicAdd(o+0, w.x*h.x); unsafeAtomicAdd(o+1, w.x*h.y);
        unsafeAtomicAdd(o+2, w.x*h.z); unsafeAtomicAdd(o+3, w.x*h.w);
        h = g[s.y]; o = t + (size_t)d.y * 4;
        unsafeAtomicAdd(o+0, w.y*h.x); unsafeAtomicAdd(o+1, w.y*h.y);
        unsafeAtomicAdd(o+2, w.y*h.z); unsafeAtomicAdd(o+3, w.y*h.w);
        h = g[s.z]; o = t + (size_t)d.z * 4;
        unsafeAtomicAdd(o+0, w.z*h.x); unsafeAtomicAdd(o+1, w.z*h.y);
        unsafeAtomicAdd(o+2, w.z*h.z); unsafeAtomicAdd(o+3, w.z*h.w);
        h = g[s.w]; o = t + (size_t)d.w * 4;
        unsafeAtomicAdd(o+0, w.w*h.x); unsafeAtomicAdd(o+1, w.w*h.y);
        unsafeAtomicAdd(o+2, w.w*h.z); unsafeAtomicAdd(o+3, w.w*h.w);
    }
}
__global__ void hop4_kernel1(const int* __restrict__ src, const int* __restrict__ dst,
                             const float* __restrict__ w, const v4f* __restrict__ g,
                             float* __restrict__ t, int E) {
    int e = blockIdx.x * blockDim.x + threadIdx.x;
    if (e < E) {
        int s = NT_LOAD(src + e), d = NT_LOAD(dst + e);
        float wv = NT_LOAD(w + e);
        v4f h = g[s];
        float* o = t + (size_t)d * 4;
        unsafeAtomicAdd(o+0, wv*h.x); unsafeAtomicAdd(o+1, wv*h.y);
        unsafeAtomicAdd(o+2, wv*h.z); unsafeAtomicAdd(o+3, wv*h.w);
    }
}

// ============== node elementwise passes (all L2-resident arrays) =============
__global__ void scale1_kernel(const float* __restrict__ dis, const float* __restrict__ x,
                              float* __restrict__ g, int N) {      // g = dis * x
    int i = blockIdx.x * blockDim.x + threadIdx.x;
    if (i < N) g[i] = dis[i] * x[i];
}
__global__ void scale1sq_kernel(const float* __restrict__ dis, const float* __restrict__ s,
                                float* __restrict__ g, int N) {    // g = dis^2 * s
    int i = blockIdx.x * blockDim.x + threadIdx.x;
    if (i < N) { float d = dis[i]; g[i] = d * d * s[i]; }
}
__global__ void scale4_kernel(const float* __restrict__ dis, const v4f* __restrict__ y,
                              v4f* __restrict__ g, int N) {        // g[n,:] = dis[n]*y[n,:]
    int i = blockIdx.x * blockDim.x + threadIdx.x;
    if (i < N) g[i] = dis[i] * y[i];
}
__global__ void scale4sq_kernel(const float* __restrict__ dis, const v4f* __restrict__ t,
                                v4f* __restrict__ g, int N) {      // g[n,:] = dis[n]^2*t[n,:]
    int i = blockIdx.x * blockDim.x + threadIdx.x;
    if (i < N) { float d = dis[i]; g[i] = (d * d) * t[i]; }
}

// ==== Layer-1 combine via V_WMMA_F32_16X16X4_F32: y1 = relu(Hcat @ W1 + b1) ====
// A[i][k]: k=0 -> x[i]; k>=1 -> dis[i]*s_k[i].  B[k][j] = w1[k][0][j&3] (cols replicated)
__global__ void combine1_wmma(const float* __restrict__ x,  const float* __restrict__ s1,
                              const float* __restrict__ s2, const float* __restrict__ s3,
                              const float* __restrict__ dis,
                              const float* __restrict__ w1, const float* __restrict__ b1,
                              float* __restrict__ y1, int N) {
    int lane = threadIdx.x & 31;
    int tile = blockIdx.x * (blockDim.x >> 5) + (threadIdx.x >> 5);
    int base = tile * 16;
    if (base >= N) return;                 // wave-uniform: EXEC all-1s for WMMA
    int m  = lane & 15;
    int hi = lane >> 4;
    int node = base + m; if (node > N - 1) node = N - 1;   // clamp tail loads
    float dn = dis[node];

    v2f a;                                  // A: VGPR0=K(2*hi), VGPR1=K(2*hi+1)
    a.x = hi ? dn * s2[node] : x[node];
    a.y = hi ? dn * s3[node] : dn * s1[node];
    v2f b;                                  // B: rows 2*hi,2*hi+1 striped over lanes
    int j = m & 3;
    b.x = w1[(2 * hi + 0) * 4 + j];
    b.y = w1[(2 * hi + 1) * 4 + j];

    v8f c = {};
    c = __builtin_amdgcn_wmma_f32_16x16x4_f32(false, a, false, b, (short)0, c, false, false);

    // D: VGPR r in lane L holds (M = r + 8*hi, N = lane&15); features in cols 0..3
    if (m < 4) {
        float bias = b1[m];
        float* yb = y1 + (size_t)(base + 8 * hi) * 4 + m;
        if (base + 16 <= N) {               // full tile: straight-line stores
            #pragma unroll
            for (int r = 0; r < 8; ++r) {
                float v = c[r] + bias;
                yb[(size_t)r * 4] = v > 0.f ? v : 0.f;
            }
        } else {
            #pragma unroll
            for (int r = 0; r < 8; ++r) {
                int nm = base + r + 8 * hi;
                if (nm < N) {
                    float v = c[r] + bias;
                    y1[(size_t)nm * 4 + m] = v > 0.f ? v : 0.f;
                }
            }
        }
    }
}

// ==== Layer-2 combine: 4 chained WMMAs (Gcat[N,16] @ w2flat[16,1]) + head ====
// G0 = y1;  G_k = dis * t_k (k>=1).  out = sigmoid((acc + b2)*wl + bl)
__global__ void combine2_wmma(const float* __restrict__ y1, const float* __restrict__ t1,
                              const float* __restrict__ t2, const float* __restrict__ t3,
                              const float* __restrict__ dis,
                              const float* __restrict__ w2, const float* __restrict__ b2,
                              const float* __restrict__ wl, const float* __restrict__ bl,
                              float* __restrict__ out, int N) {
    int lane = threadIdx.x & 31;
    int tile = blockIdx.x * (blockDim.x >> 5) + (threadIdx.x >> 5);
    int base = tile * 16;
    if (base >= N) return;
    int m  = lane & 15;
    int hi = lane >> 4;
    int node = base + m; if (node > N - 1) node = N - 1;
    size_t roff = (size_t)node * 4 + 2 * hi;
    float dn = dis[node];

    v8f c = {};
    v2f a, b;
    // k = 0 (unscaled y1)
    a.x = y1[roff]; a.y = y1[roff + 1];
    b.x = w2[0 + 2 * hi]; b.y = w2[0 + 2 * hi + 1];
    c = __builtin_amdgcn_wmma_f32_16x16x4_f32(false, a, false, b, (short)0, c, false, false);
    // k = 1
    a.x = dn * t1[roff]; a.y = dn * t1[roff + 1];
    b.x = w2[4 + 2 * hi]; b.y = w2[4 + 2 * hi + 1];
    c = __builtin_amdgcn_wmma_f32_16x16x4_f32(false, a, false, b, (short)0, c, false, false);
    // k = 2
    a.x = dn * t2[roff]; a.y = dn * t2[roff + 1];
    b.x = w2[8 + 2 * hi]; b.y = w2[8 + 2 * hi + 1];
    c = __builtin_amdgcn_wmma_f32_16x16x4_f32(false, a, false, b, (short)0, c, false, false);
    // k = 3
    a.x = dn * t3[roff]; a.y = dn * t3[roff + 1];
    b.x = w2[12 + 2 * hi]; b.y = w2[12 + 2 * hi + 1];
    c = __builtin_amdgcn_wmma_f32_16x16x4_f32(false, a, false, b, (short)0, c, false, false);

    // All 16 D columns identical (B uniform within lane-half) -> use column 0
    if (m == 0) {
        float bb = b2[0], wll = wl[0], bll = bl[0];
        float* ob = out + base + 8 * hi;
        if (base + 16 <= N) {
            #pragma unroll
            for (int r = 0; r < 8; ++r) {
                float v = (c[r] + bb) * wll + bll;
                ob[r] = 1.f / (1.f + expf(-v));
            }
        } else {
            #pragma unroll
            for (int r = 0; r < 8; ++r) {
                int nm = base + r + 8 * hi;
                if (nm < N) {
                    float v = (c[r] + bb) * wll + bll;
                    out[nm] = 1.f / (1.f + expf(-v));
                }
            }
        }
    }
}

extern "C" void kernel_launch(void* const* d_in, const int* in_sizes, int n_in,
                              void* d_out, int out_size, void* d_ws, size_t ws_size,
                              hipStream_t stream) {
    const float* x   = (const float*)d_in[0];          // [N,1]
    const int*   ei  = (const int*)d_in[1];            // [2,E]
    const float* ew  = (const float*)d_in[2];          // [E]
    const float* w1  = (const float*)d_in[3];          // [4,1,4]
    const float* b1  = (const float*)d_in[4];          // [4]
    const float* w2  = (const float*)d_in[5];          // [4,4,1]
    const float* b2  = (const float*)d_in[6];          // [1]
    const float* wl  = (const float*)d_in[7];          // [1,1]
    const float* bl  = (const float*)d_in[8];          // [1]
    float* out = (float*)d_out;

    const int N = in_sizes[0];
    const int E = in_sizes[2];
    const int* src = ei;
    const int* dst = ei + E;

    // Workspace (floats):
    // [ dis N | s1 N | s2 N | s3 N | t1 4N | t2 4N | t3 4N ]  <- zeroed (16N)
    // [ y1 4N | gA N | gB N | qA 4N | qB 4N ]
    float* ws  = (float*)d_ws;
    float* dis = ws;                        // deg then dis (in place)
    float* s1  = ws + (size_t)N;
    float* s2  = ws + (size_t)2  * N;
    float* s3  = ws + (size_t)3  * N;
    float* t1  = ws + (size_t)4  * N;
    float* t2  = ws + (size_t)8  * N;
    float* t3  = ws + (size_t)12 * N;
    float* y1  = ws + (size_t)16 * N;
    float* gA  = ws + (size_t)20 * N;
    float* gB  = ws + (size_t)21 * N;
    float* qA  = ws + (size_t)22 * N;
    float* qB  = ws + (size_t)26 * N;

    hipMemsetAsync(ws, 0, (size_t)16 * N * sizeof(float), stream);

    const int BT = 256;
    const bool quad = ((E & 3) == 0);       // v4 path needs 16B-aligned dst = ei+E
    const int Q   = quad ? (E >> 2) : 0;
    dim3 qg(quad ? (Q + BT - 1) / BT : 1), qb(BT);
    dim3 eg((E + BT - 1) / BT), eb(BT);
    dim3 ng((N + BT - 1) / BT), nb(BT);
    dim3 cg((N + 127) / 128), cb(256);      // 8 waves/block, 16 nodes/wave

    const v4i* src4 = (const v4i*)src;
    const v4i* dst4 = (const v4i*)dst;
    const v4f* ew4  = (const v4f*)ew;

    // ---- gcn_norm ----
    if (quad) deg_kernel4<<<qg, qb, 0, stream>>>(dst4, ew4, dis, Q);
    else      deg_kernel1<<<eg, eb, 0, stream>>>(dst, ew, dis, E);
    rsqrt_kernel<<<ng, nb, 0, stream>>>(dis, N);

    // ---- layer 1: three F=1 hops on dis-prescaled g, s_k = W g_{k-1} ----
    scale1_kernel<<<ng, nb, 0, stream>>>(dis, x, gA, N);            // g0 = dis*x
    if (quad) hop1_kernel4<<<qg, qb, 0, stream>>>(src4, dst4, ew4, gA, s1, Q);
    else      hop1_kernel1<<<eg, eb, 0, stream>>>(src, dst, ew, gA, s1, E);
    scale1sq_kernel<<<ng, nb, 0, stream>>>(dis, s1, gB, N);         // g1 = dis^2*s1
    if (quad) hop1_kernel4<<<qg, qb, 0, stream>>>(src4, dst4, ew4, gB, s2, Q);
    else      hop1_kernel1<<<eg, eb, 0, stream>>>(src, dst, ew, gB, s2, E);
    scale1sq_kernel<<<ng, nb, 0, stream>>>(dis, s2, gA, N);         // g2 = dis^2*s2
    if (quad) hop1_kernel4<<<qg, qb, 0, stream>>>(src4, dst4, ew4, gA, s3, Q);
    else      hop1_kernel1<<<eg, eb, 0, stream>>>(src, dst, ew, gA, s3, E);
    combine1_wmma<<<cg, cb, 0, stream>>>(x, s1, s2, s3, dis, w1, b1, y1, N);

    // ---- layer 2: three F=4 hops, same prescale trick ----
    scale4_kernel<<<ng, nb, 0, stream>>>(dis, (const v4f*)y1, (v4f*)qA, N);
    if (quad) hop4_kernel4<<<qg, qb, 0, stream>>>(src4, dst4, ew4, (const v4f*)qA, t1, Q);
    else      hop4_kernel1<<<eg, eb, 0, stream>>>(src, dst, ew, (const v4f*)qA, t1, E);
    scale4sq_kernel<<<ng, nb, 0, stream>>>(dis, (const v4f*)t1, (v4f*)qB, N);
    if (quad) hop4_kernel4<<<qg, qb, 0, stream>>>(src4, dst4, ew4, (const v4f*)qB, t2, Q);
    else      hop4_kernel1<<<eg, eb, 0, stream>>>(src, dst, ew, (const v4f*)qB, t2, E);
    scale4sq_kernel<<<ng, nb, 0, stream>>>(dis, (const v4f*)t2, (v4f*)qA, N);
    if (quad) hop4_kernel4<<<qg, qb, 0, stream>>>(src4, dst4, ew4, (const v4f*)qA, t3, Q);
    else      hop4_kernel1<<<eg, eb, 0, stream>>>(src, dst, ew, (const v4f*)qA, t3, E);
    combine2_wmma<<<cg, cb, 0, stream>>>(y1, t1, t2, t3, dis, w2, b2, wl, bl, out, N);
}